// TransformerBlockQuantum_65481071400592
// MI455X (gfx1250) — compile-verified
//
#include <hip/hip_runtime.h>

// CDNA5 / gfx1250, wave32. Fused quantum-transformer block.
// fp32 WMMA (V_WMMA_F32_16X16X4_F32): K=4 matches DK=4 (attention) and E=8=2*K (FFN GEMM1).
// Flash-style attention: the 536MB score tensor is never materialized.
// Cross-lane reductions use DPP butterflies (pure VALU) instead of ds_bpermute.
// Softmax max-subtraction dropped: scores provably bounded in [-2,2] (cos inputs),
// so exp never overflows and per-lane partials can accumulate across all chunks,
// with a single DPP reduction per row at the end.

typedef __attribute__((ext_vector_type(2))) float v2f;
typedef __attribute__((ext_vector_type(8))) float v8f;

#define NB 16
#define NS 2048
#define NE 8
#define NF 2048
#define LN_EPS 1e-5f

// DPP controls: 0xB1 = quad_perm [1,0,3,2] (xor1), 0x4E = quad_perm [2,3,0,1] (xor2),
// 0x141 = row_half_mirror (other quad of octet), 0x140 = row_mirror (other octet of row16).
template <int CTRL>
static __device__ __forceinline__ float dppf(float v) {
  return __int_as_float(__builtin_amdgcn_update_dpp(
      0, __float_as_int(v), CTRL, 0xF, 0xF, true));
}
// sum across the 16 lanes of a half-wave (DPP rows are 16 lanes)
static __device__ __forceinline__ float xsum16(float v) {
  v += dppf<0xB1>(v);
  v += dppf<0x4E>(v);
  v += dppf<0x141>(v);
  v += dppf<0x140>(v);
  return v;
}
// sum across an 8-lane octet
static __device__ __forceinline__ float xsum8(float v) {
  v += dppf<0xB1>(v);
  v += dppf<0x4E>(v);
  v += dppf<0x141>(v);
  return v;
}

// ---------------------------------------------------------------------------
// Kernel 1: flash attention per (b, h, 16-query tile). 4 waves / block.
// qx = cos(x + theta). Scores tile via V_WMMA_F32_16X16X4_F32 (K = DK = 4).
// Per-lane partial softmax sums accumulate across all key chunks; one DPP
// reduction per row at the end. ao[b,s,4h..4h+3] = per-head attention output.
// ---------------------------------------------------------------------------
__global__ __launch_bounds__(128) void qattn_flash_kernel(
    const float* __restrict__ x, const float* __restrict__ theta,
    float* __restrict__ ao) {
  const int wid  = blockIdx.x * 4 + (threadIdx.x >> 5);
  const int lane = threadIdx.x & 31;
  const int hw   = lane >> 4;   // which 16-lane half
  const int lm   = lane & 15;
  const int b  = wid >> 8;        // 256 waves per batch (H * S/16)
  const int h  = (wid >> 7) & 1;  // head
  const int q0 = (wid & 127) << 4;

  const float4 th = *reinterpret_cast<const float4*>(theta + 4 * h);
  const float thk0 = hw ? th.z : th.x;  // theta[4h + 2*hw + 0]
  const float thk1 = hw ? th.w : th.y;  // theta[4h + 2*hw + 1]

  // A tile (16 queries x 4): lane holds A[lm][2*hw+v]
  v2f amat;
  {
    const float* qrow = x + ((size_t)b * NS + q0 + lm) * NE + 4 * h;
    const float2 q2 = *reinterpret_cast<const float2*>(qrow + 2 * hw);
    amat[0] = __cosf(q2.x + thk0);
    amat[1] = __cosf(q2.y + thk1);
  }

  // per-lane partial sums: lacc = sum_p, pv[d] = sum_p * v_d (this lane's cols)
  float lacc[8], pv[8][4];
#pragma unroll
  for (int r = 0; r < 8; ++r) {
    lacc[r] = 0.f;
    pv[r][0] = pv[r][1] = pv[r][2] = pv[r][3] = 0.f;
  }

  const float* xk = x + (size_t)b * NS * NE + 4 * h;
  for (int kc = 0; kc < NS; kc += 16) {
    // key/value row for this lane's column: all 4 head dims in one float4
    float4 vv = *reinterpret_cast<const float4*>(xk + (size_t)(kc + lm) * NE);
    vv.x = __cosf(vv.x + th.x);
    vv.y = __cosf(vv.y + th.y);
    vv.z = __cosf(vv.z + th.z);
    vv.w = __cosf(vv.w + th.w);
    // B tile (4 x 16 keys): lane holds B[2*hw+v][lm] = K[col lm][dim 2*hw+v]
    v2f bmat;
    bmat[0] = hw ? vv.z : vv.x;
    bmat[1] = hw ? vv.w : vv.y;

    v8f z = {0.f, 0.f, 0.f, 0.f, 0.f, 0.f, 0.f, 0.f};
    v8f s = __builtin_amdgcn_wmma_f32_16x16x4_f32(false, amat, false, bmat,
                                                  (short)0, z, false, false);
    // D layout: VGPR r holds score[row r+8*hw][col lm] for this 16-key chunk.
    // |score*0.5| <= 2 (cos-bounded), so plain exp is numerically safe.
#pragma unroll
    for (int r = 0; r < 8; ++r) {
      const float p = __expf(s[r] * 0.5f);   // 0.5 = 1/sqrt(DK)
      lacc[r] += p;
      pv[r][0] = fmaf(p, vv.x, pv[r][0]);
      pv[r][1] = fmaf(p, vv.y, pv[r][1]);
      pv[r][2] = fmaf(p, vv.z, pv[r][2]);
      pv[r][3] = fmaf(p, vv.w, pv[r][3]);
    }
  }

  // One reduction per row at the very end; every lane of the half then holds
  // the row result, lane 0 stores rows 0..7, lane 16 stores rows 8..15.
#pragma unroll
  for (int r = 0; r < 8; ++r) {
    const float inv = 1.0f / xsum16(lacc[r]);
    float4 o;
    o.x = xsum16(pv[r][0]) * inv;
    o.y = xsum16(pv[r][1]) * inv;
    o.z = xsum16(pv[r][2]) * inv;
    o.w = xsum16(pv[r][3]) * inv;
    if (lm == 0) {
      *reinterpret_cast<float4*>(
          ao + ((size_t)b * NS + q0 + r + 8 * hw) * NE + 4 * h) = o;
    }
  }
}

// ---------------------------------------------------------------------------
// Kernel 2: attn_out = ao @ w_out^T (8x8, uniform -> s_loads), residual, LN1.
// One token per lane. Writes x1 into d_out (consumed in-place by kernel 3).
// ---------------------------------------------------------------------------
__global__ __launch_bounds__(256) void proj_ln1_kernel(
    const float* __restrict__ x, const float* __restrict__ ao,
    const float* __restrict__ w_out, const float* __restrict__ g,
    const float* __restrict__ be, float* __restrict__ x1) {
  const int t = blockIdx.x * 256 + threadIdx.x;
  if (t >= NB * NS) return;

  float av[8], xv[8], tv[8];
  {
    const float4 a0 = *reinterpret_cast<const float4*>(ao + (size_t)t * NE);
    const float4 a1 = *reinterpret_cast<const float4*>(ao + (size_t)t * NE + 4);
    av[0] = a0.x; av[1] = a0.y; av[2] = a0.z; av[3] = a0.w;
    av[4] = a1.x; av[5] = a1.y; av[6] = a1.z; av[7] = a1.w;
    const float4 x0 = *reinterpret_cast<const float4*>(x + (size_t)t * NE);
    const float4 x4 = *reinterpret_cast<const float4*>(x + (size_t)t * NE + 4);
    xv[0] = x0.x; xv[1] = x0.y; xv[2] = x0.z; xv[3] = x0.w;
    xv[4] = x4.x; xv[5] = x4.y; xv[6] = x4.z; xv[7] = x4.w;
  }
  float mean = 0.f;
#pragma unroll
  for (int e = 0; e < 8; ++e) {
    float s = xv[e];
#pragma unroll
    for (int i = 0; i < 8; ++i) s += av[i] * w_out[e * 8 + i];
    tv[e] = s;
    mean += s;
  }
  mean *= 0.125f;
  float var = 0.f;
#pragma unroll
  for (int e = 0; e < 8; ++e) {
    const float d = tv[e] - mean;
    var += d * d;
  }
  var *= 0.125f;
  const float rs = rsqrtf(var + LN_EPS);
  float4 o0, o1;
  o0.x = (tv[0] - mean) * rs * g[0] + be[0];
  o0.y = (tv[1] - mean) * rs * g[1] + be[1];
  o0.z = (tv[2] - mean) * rs * g[2] + be[2];
  o0.w = (tv[3] - mean) * rs * g[3] + be[3];
  o1.x = (tv[4] - mean) * rs * g[4] + be[4];
  o1.y = (tv[5] - mean) * rs * g[5] + be[5];
  o1.z = (tv[6] - mean) * rs * g[6] + be[6];
  o1.w = (tv[7] - mean) * rs * g[7] + be[7];
  *reinterpret_cast<float4*>(x1 + (size_t)t * NE)     = o0;
  *reinterpret_cast<float4*>(x1 + (size_t)t * NE + 4) = o1;
}

// ---------------------------------------------------------------------------
// Kernel 3: fused FFN (8 -> 2048 -> relu -> 8) + residual + LN2, per 16-token
// tile (one wave each, 4 waves/block). GEMM1: 2 chained K=4 fp32 WMMAs.
// relu'd 16x16 tile: D-layout -> LDS -> A-layout, then GEMM2: 4 chained WMMAs
// per 16-wide FFN chunk (N=8 zero-padded to 16). x1 is read/written in d_out;
// each tile only touches its own rows, loads precede stores per row.
// ---------------------------------------------------------------------------
__global__ __launch_bounds__(128) void ffn_ln2_kernel(
    const float* x1, const float* __restrict__ phi,
    const float* __restrict__ w1, const float* __restrict__ w2,
    const float* __restrict__ g, const float* __restrict__ be, float* out) {
  __shared__ float pbuf[4][256];
  const int wv   = threadIdx.x >> 5;
  const int lane = threadIdx.x & 31;
  const int hw   = lane >> 4;
  const int lm   = lane & 15;
  const int tok0 = (blockIdx.x * 4 + wv) * 16;

  // A tiles: qf = cos(x1)*cos(phi). A0 covers K=0..3, A1 covers K=4..7.
  v2f a0, a1;
  {
    const float* xr = x1 + (size_t)(tok0 + lm) * NE;
    const float2 xa  = *reinterpret_cast<const float2*>(xr + 2 * hw);
    const float2 xb  = *reinterpret_cast<const float2*>(xr + 4 + 2 * hw);
    const float2 p0  = *reinterpret_cast<const float2*>(phi + 2 * hw);
    const float2 p1  = *reinterpret_cast<const float2*>(phi + 4 + 2 * hw);
    a0[0] = __cosf(xa.x) * __cosf(p0.x);
    a0[1] = __cosf(xa.y) * __cosf(p0.y);
    a1[0] = __cosf(xb.x) * __cosf(p1.x);
    a1[1] = __cosf(xb.y) * __cosf(p1.y);
  }

  const float nmask = (lm < 8) ? 1.0f : 0.0f;  // zero-pad N=8 -> 16 (no divergence)
  v8f acc = {0.f, 0.f, 0.f, 0.f, 0.f, 0.f, 0.f, 0.f};

  for (int f0 = 0; f0 < NF; f0 += 16) {
    // GEMM1: D1[16t x 16f] = qf[16 x 8] @ w1chunk[8 x 16]
    const float* w1r = w1 + (size_t)(f0 + lm) * NE;
    const float2 b0f = *reinterpret_cast<const float2*>(w1r + 2 * hw);
    const float2 b1f = *reinterpret_cast<const float2*>(w1r + 4 + 2 * hw);
    v2f b0; b0[0] = b0f.x; b0[1] = b0f.y;
    v2f b1; b1[0] = b1f.x; b1[1] = b1f.y;
    v8f z = {0.f, 0.f, 0.f, 0.f, 0.f, 0.f, 0.f, 0.f};
    v8f d1 = __builtin_amdgcn_wmma_f32_16x16x4_f32(false, a0, false, b0,
                                                   (short)0, z, false, false);
    d1 = __builtin_amdgcn_wmma_f32_16x16x4_f32(false, a1, false, b1,
                                               (short)0, d1, false, false);
    // relu, spill D-layout tile to LDS (row-major 16x16), reload as A-layout
#pragma unroll
    for (int r = 0; r < 8; ++r)
      pbuf[wv][(r + 8 * hw) * 16 + lm] = fmaxf(d1[r], 0.0f);
#pragma unroll
    for (int j = 0; j < 4; ++j) {
      const float2 pa =
          *reinterpret_cast<const float2*>(&pbuf[wv][lm * 16 + 4 * j + 2 * hw]);
      v2f av; av[0] = pa.x; av[1] = pa.y;
      // B: w2chunk[4f x 16], cols 0..7 = embed dims, cols 8..15 zero
      const float2 bw2 = *reinterpret_cast<const float2*>(
          w2 + (size_t)(lm & 7) * NF + f0 + 4 * j + 2 * hw);
      v2f bw; bw[0] = bw2.x * nmask; bw[1] = bw2.y * nmask;
      acc = __builtin_amdgcn_wmma_f32_16x16x4_f32(false, av, false, bw,
                                                  (short)0, acc, false, false);
    }
  }

  // residual + LN2. Row m's 8 valid values live in lanes 0..7 of a half.
#pragma unroll
  for (int r = 0; r < 8; ++r) {
    const int row = tok0 + r + 8 * hw;
    const float val = acc[r] + x1[(size_t)row * NE + (lm & 7)];
    const float mean = xsum8(val) * 0.125f;
    const float var  = xsum8(val * val) * 0.125f - mean * mean;
    const float y = (val - mean) * rsqrtf(var + LN_EPS) * g[lm & 7] + be[lm & 7];
    if (lm < 8) out[(size_t)row * NE + lm] = y;
  }
}

extern "C" void kernel_launch(void* const* d_in, const int* in_sizes, int n_in,
                              void* d_out, int out_size, void* d_ws, size_t ws_size,
                              hipStream_t stream) {
  (void)in_sizes; (void)n_in; (void)out_size; (void)ws_size;
  const float* x     = (const float*)d_in[0];
  const float* theta = (const float*)d_in[1];
  const float* w_out = (const float*)d_in[2];
  const float* ln1_g = (const float*)d_in[3];
  const float* ln1_b = (const float*)d_in[4];
  const float* phi   = (const float*)d_in[5];
  const float* w1    = (const float*)d_in[6];
  const float* w2    = (const float*)d_in[7];
  const float* ln2_g = (const float*)d_in[8];
  const float* ln2_b = (const float*)d_in[9];
  float* out = (float*)d_out;
  float* ao  = (float*)d_ws;  // B*S*E floats = 1 MiB attention-output scratch

  // 16*2*128 = 4096 query-tile waves, 4 per block
  qattn_flash_kernel<<<1024, 128, 0, stream>>>(x, theta, ao);
  // 32768 tokens, 256 per block
  proj_ln1_kernel<<<128, 256, 0, stream>>>(x, ao, w_out, ln1_g, ln1_b, out);
  // 2048 16-token tiles, 4 waves per block (x1 lives in d_out, updated in place)
  ffn_ln2_kernel<<<512, 128, 0, stream>>>(out, phi, w1, w2, ln2_g, ln2_b, out);
}